// CHRONOSNet_25005299597837
// MI455X (gfx1250) — compile-verified
//
#include <hip/hip_runtime.h>
#include <hip/hip_bf16.h>
#include <math.h>

typedef float v2f __attribute__((ext_vector_type(2)));
typedef float v8f __attribute__((ext_vector_type(8)));

#define NEG_SLOPE 0.2f
#define BN_EPS 1e-5f

// ---------------------------------------------------------------------------
// Fill kernel (workspace init; harness poisons buffers, we must zero/-inf them)
// ---------------------------------------------------------------------------
__global__ void fill_f32(float* __restrict__ p, float v, long n) {
  long t = (long)blockIdx.x * blockDim.x + threadIdx.x;
  long stride = (long)gridDim.x * blockDim.x;
  for (; t < n; t += stride) p[t] = v;
}

// ---------------------------------------------------------------------------
// FP32 WMMA GEMM: C[M,Nc] = A[M,K] @ B[K,Nc] (+bias, optional relu)
// One wave -> 16x64 output tile (4 x v8f accumulators), K stepped by 4 using
// V_WMMA_F32_16X16X4_F32. Requires M%16==0, Nc%64==0, K%4==0 (true for all
// GEMMs in this net: M=20000, Nc in {256,2048,128}, K in {236,256,512}).
// ---------------------------------------------------------------------------
__global__ void wmma_gemm_f32(const float* __restrict__ A,
                              const float* __restrict__ B,
                              const float* __restrict__ bias,
                              float* __restrict__ C,
                              int M, int K, int Nc, int relu) {
  const int wavesPerBlock = blockDim.x >> 5;
  const int wave = blockIdx.x * wavesPerBlock + (threadIdx.x >> 5);
  const int lane = threadIdx.x & 31;
  const int mtiles = M >> 4;
  const int ntiles = Nc >> 6;
  if (wave >= mtiles * ntiles) return;   // whole-wave uniform exit
  const int mt = wave / ntiles;
  const int nt = wave % ntiles;
  const int half = lane >> 4;            // 0: K pair {0,1}, 1: K pair {2,3}
  const int l16  = lane & 15;
  const int row  = mt * 16 + l16;        // A-fragment row for this lane
  const int nbase = nt * 64;

  v8f acc0 = {}, acc1 = {}, acc2 = {}, acc3 = {};
  const float* __restrict__ Arow = A + (size_t)row * K;

  for (int kk = 0; kk < K; kk += 4) {
    // A 16x4 fragment: lanes 0-15 hold K={kk,kk+1}, lanes 16-31 K={kk+2,kk+3}
    v2f a;
    a.x = Arow[kk + 2 * half];
    a.y = Arow[kk + 2 * half + 1];
    const float* __restrict__ B0 = B + (size_t)(kk + 2 * half) * Nc;
    const float* __restrict__ B1 = B + (size_t)(kk + 2 * half + 1) * Nc;
    // 4 adjacent 4x16 B fragments share the A fragment
    {
      v2f b; int col = nbase + 0 * 16 + l16;
      b.x = B0[col]; b.y = B1[col];
      acc0 = __builtin_amdgcn_wmma_f32_16x16x4_f32(false, a, false, b, (short)0, acc0, false, false);
    }
    {
      v2f b; int col = nbase + 1 * 16 + l16;
      b.x = B0[col]; b.y = B1[col];
      acc1 = __builtin_amdgcn_wmma_f32_16x16x4_f32(false, a, false, b, (short)0, acc1, false, false);
    }
    {
      v2f b; int col = nbase + 2 * 16 + l16;
      b.x = B0[col]; b.y = B1[col];
      acc2 = __builtin_amdgcn_wmma_f32_16x16x4_f32(false, a, false, b, (short)0, acc2, false, false);
    }
    {
      v2f b; int col = nbase + 3 * 16 + l16;
      b.x = B0[col]; b.y = B1[col];
      acc3 = __builtin_amdgcn_wmma_f32_16x16x4_f32(false, a, false, b, (short)0, acc3, false, false);
    }
  }

  // C/D layout: lanes 0-15 -> M = vgpr index, lanes 16-31 -> M = 8 + vgpr
  v8f accs[4] = {acc0, acc1, acc2, acc3};
#pragma unroll
  for (int s = 0; s < 4; ++s) {
    const int col = nbase + s * 16 + l16;
    const float bv = bias ? bias[col] : 0.0f;
#pragma unroll
    for (int i = 0; i < 8; ++i) {
      const int r = mt * 16 + half * 8 + i;
      float v = accs[s][i] + bv;
      if (relu) v = v > 0.0f ? v : 0.0f;
      C[(size_t)r * Nc + col] = v;
    }
  }
}

// ---------------------------------------------------------------------------
// Attention logits: als[n,h] = sum_c x[n,h,c]*a_s[h,c]; same for ald.
// One wave per (n,h), lanes stride over channels, shuffle reduction.
// ---------------------------------------------------------------------------
__global__ void attn_logits(const float* __restrict__ x,
                            const float* __restrict__ a_s,
                            const float* __restrict__ a_d,
                            float* __restrict__ als, float* __restrict__ ald,
                            int N, int H, int C) {
  const int lane = threadIdx.x & 31;
  const long w = (long)blockIdx.x * (blockDim.x >> 5) + (threadIdx.x >> 5);
  if (w >= (long)N * H) return;
  const int h = (int)(w % H);
  const long n = w / H;
  const float* __restrict__ xr = x + (size_t)n * H * C + (size_t)h * C;
  const float* __restrict__ pas = a_s + (size_t)h * C;
  const float* __restrict__ pad = a_d + (size_t)h * C;
  float ss = 0.0f, sd = 0.0f;
  for (int c = lane; c < C; c += 32) {
    const float xv = xr[c];
    ss += xv * pas[c];
    sd += xv * pad[c];
  }
#pragma unroll
  for (int o = 16; o; o >>= 1) {
    ss += __shfl_xor(ss, o, 32);
    sd += __shfl_xor(sd, o, 32);
  }
  if (lane == 0) { als[w] = ss; ald[w] = sd; }
}

// ---------------------------------------------------------------------------
// Float atomic max via int/uint ordering trick
// ---------------------------------------------------------------------------
__device__ __forceinline__ void atomicMaxF(float* addr, float val) {
  if (val >= 0.0f) atomicMax((int*)addr, __float_as_int(val));
  else             atomicMin((unsigned int*)addr, (unsigned int)__float_as_int(val));
}

// Edge pass 1: e = leaky_relu(als[src]+ald[dst]); segment max into m[dst]
__global__ void edge_logit_max(const int* __restrict__ ei, int E, int N, int H,
                               const float* __restrict__ als,
                               const float* __restrict__ ald,
                               float* __restrict__ ebuf, float* __restrict__ m) {
  const long t = (long)blockIdx.x * blockDim.x + threadIdx.x;
  const long tot = (long)(E + N) * H;
  if (t >= tot) return;
  const int h = (int)(t % H);
  const long e = t / H;
  const int s = (e < E) ? ei[e]     : (int)(e - E);
  const int d = (e < E) ? ei[E + e] : (int)(e - E);
  float v = als[(size_t)s * H + h] + ald[(size_t)d * H + h];
  v = v > 0.0f ? v : NEG_SLOPE * v;
  ebuf[t] = v;
  atomicMaxF(&m[(size_t)d * H + h], v);
}

// Edge pass 2: ex = exp(e - m[dst]); den[dst] += ex
__global__ void edge_exp_sum(const int* __restrict__ ei, int E, int N, int H,
                             const float* __restrict__ m,
                             float* __restrict__ ebuf, float* __restrict__ den) {
  const long t = (long)blockIdx.x * blockDim.x + threadIdx.x;
  const long tot = (long)(E + N) * H;
  if (t >= tot) return;
  const int h = (int)(t % H);
  const long e = t / H;
  const int d = (e < E) ? ei[E + e] : (int)(e - E);
  const float ex = expf(ebuf[t] - m[(size_t)d * H + h]);
  ebuf[t] = ex;
  atomicAdd(&den[(size_t)d * H + h], ex);
}

// Edge pass 3: agg[dst] += alpha * x[src]  (optionally averaged over heads)
__global__ void edge_aggregate(const int* __restrict__ ei, int E, int N, int H, int C,
                               const float* __restrict__ x,
                               const float* __restrict__ ebuf,
                               const float* __restrict__ den,
                               float* __restrict__ agg, int mean_heads) {
  const long tot = (long)(E + N) * H * C;
  const long stride = (long)gridDim.x * blockDim.x;
  const float hscale = 1.0f / (float)H;
  for (long t = (long)blockIdx.x * blockDim.x + threadIdx.x; t < tot; t += stride) {
    const int c = (int)(t % C);
    const long eh = t / C;
    const int h = (int)(eh % H);
    const long e = eh / H;
    const int s = (e < E) ? ei[e]     : (int)(e - E);
    const int d = (e < E) ? ei[E + e] : (int)(e - E);
    const float alpha = ebuf[eh] / (den[(size_t)d * H + h] + 1e-16f);
    const float v = alpha * x[(size_t)s * H * C + (size_t)h * C + c];
    if (mean_heads) atomicAdd(&agg[(size_t)d * C + c], v * hscale);
    else            atomicAdd(&agg[(size_t)d * H * C + (size_t)h * C + c], v);
  }
}

// ---------------------------------------------------------------------------
// BatchNorm stats: per-channel sum & sumsq of (X + bias). blockDim.x == C.
// ---------------------------------------------------------------------------
__global__ void bn_stats(const float* __restrict__ X, const float* __restrict__ bias,
                         float* __restrict__ stats, int N, int C) {
  const int c = threadIdx.x;
  const float b = bias[c];
  float s = 0.0f, s2 = 0.0f;
  for (int r = blockIdx.x; r < N; r += gridDim.x) {
    const float v = X[(size_t)r * C + c] + b;
    s += v; s2 += v * v;
  }
  atomicAdd(&stats[c], s);
  atomicAdd(&stats[C + c], s2);
}

// BN apply (+bias), optional ELU + residual
__global__ void bn_apply(const float* __restrict__ X, const float* __restrict__ bias,
                         const float* __restrict__ gamma, const float* __restrict__ beta,
                         const float* __restrict__ stats, const float* __restrict__ resid,
                         float* __restrict__ out, int N, int C, int elu_res) {
  const long t = (long)blockIdx.x * blockDim.x + threadIdx.x;
  if (t >= (long)N * C) return;
  const int c = (int)(t % C);
  const float invN = 1.0f / (float)N;
  const float mu = stats[c] * invN;
  const float var = stats[C + c] * invN - mu * mu;
  float v = (X[t] + bias[c] - mu) * rsqrtf(var + BN_EPS) * gamma[c] + beta[c];
  if (elu_res) {
    v = v > 0.0f ? v : (expf(v) - 1.0f);
    v += resid[t];
  }
  out[t] = v;
}

// hc = concat([a, b], axis=1)
__global__ void concat2(const float* __restrict__ a, const float* __restrict__ b,
                        float* __restrict__ out, int N, int C) {
  const long t = (long)blockIdx.x * blockDim.x + threadIdx.x;
  if (t >= (long)N * 2 * C) return;
  const int c = (int)(t % (2 * C));
  const long r = t / (2 * C);
  out[t] = (c < C) ? a[(size_t)r * C + c] : b[(size_t)r * C + (c - C)];
}

// Final tiny FC: [N,K] @ [K,2] + b  (scalar; K=128, not WMMA-worthy)
__global__ void final_fc2(const float* __restrict__ z, const float* __restrict__ W,
                          const float* __restrict__ b, float* __restrict__ out,
                          int N, int K) {
  const int r = blockIdx.x * blockDim.x + threadIdx.x;
  if (r >= N) return;
  float a0 = b[0], a1 = b[1];
  const float* __restrict__ zr = z + (size_t)r * K;
  for (int k = 0; k < K; ++k) {
    const float v = zr[k];
    a0 += v * W[2 * k];
    a1 += v * W[2 * k + 1];
  }
  out[2 * r]     = a0;
  out[2 * r + 1] = a1;
}

// ---------------------------------------------------------------------------
// Host orchestration
// ---------------------------------------------------------------------------
static inline long cdivl(long a, long b) { return (a + b - 1) / b; }

extern "C" void kernel_launch(void* const* d_in, const int* in_sizes, int n_in,
                              void* d_out, int out_size, void* d_ws, size_t ws_size,
                              hipStream_t stream) {
  // ---- inputs (setup_inputs dict order) ----
  const float* x_in = (const float*)d_in[0];
  const int*   ei   = (const int*)  d_in[1];
  const float* Wp   = (const float*)d_in[2];
  const float* bp   = (const float*)d_in[3];
  const float* W[3]  = {(const float*)d_in[4],  (const float*)d_in[10], (const float*)d_in[16]};
  const float* As[3] = {(const float*)d_in[5],  (const float*)d_in[11], (const float*)d_in[17]};
  const float* Ad[3] = {(const float*)d_in[6],  (const float*)d_in[12], (const float*)d_in[18]};
  const float* Bb[3] = {(const float*)d_in[7],  (const float*)d_in[13], (const float*)d_in[19]};
  const float* Gg[3] = {(const float*)d_in[8],  (const float*)d_in[14], (const float*)d_in[20]};
  const float* Be[3] = {(const float*)d_in[9],  (const float*)d_in[15], (const float*)d_in[21]};
  const float* cW1 = (const float*)d_in[22]; const float* cb1 = (const float*)d_in[23];
  const float* cW2 = (const float*)d_in[24]; const float* cb2 = (const float*)d_in[25];
  const float* cW3 = (const float*)d_in[26]; const float* cb3 = (const float*)d_in[27];

  const int HID = in_sizes[3];               // 256
  const int F   = in_sizes[2] / HID;         // 236
  const int Nn  = in_sizes[0] / F;           // 20000
  const int E   = in_sizes[1] / 2;           // 160000
  const int H   = in_sizes[17] / HID;        // 8 (as2 = [H, HID])
  const int C   = HID / H;                   // 32
  const int HID2 = in_sizes[25];             // 128 (cb2)
  const int Etot = E + Nn;

  // ---- workspace carve-up (floats) ----
  float* ws = (float*)d_ws;
  size_t o = 0;
  float* h0   = ws + o; o += (size_t)Nn * HID;          // h_temporal (input proj)
  float* hA   = ws + o; o += (size_t)Nn * HID;          // ping
  float* hB   = ws + o; o += (size_t)Nn * HID;          // pong
  float* xb   = ws + o; o += (size_t)Nn * H * HID;      // per-layer transform (max N x 2048)
  float* als  = ws + o; o += (size_t)Nn * H;
  float* ald  = ws + o; o += (size_t)Nn * H;
  float* mbuf = ws + o; o += (size_t)Nn * H;
  float* den  = ws + o; o += (size_t)Nn * H;
  float* ebuf = ws + o; o += (size_t)Etot * H;
  float* agg  = ws + o; o += (size_t)Nn * HID;
  float* stats= ws + o; o += (size_t)2 * HID;
  // classifier buffers alias xb (free by then): 512+256+128 = 896 <= 2048 cols
  float* hc = xb;
  float* z1 = xb + (size_t)Nn * 2 * HID;
  float* z2 = z1 + (size_t)Nn * HID;

  const int TB = 256;
  const int wavesPerBlock = TB >> 5;

  auto launch_gemm = [&](const float* A, const float* Bm, const float* bias,
                         float* Cm, int M, int K, int Nc, int relu) {
    const long waves = (long)(M >> 4) * (Nc >> 6);
    wmma_gemm_f32<<<(int)cdivl(waves, wavesPerBlock), TB, 0, stream>>>(A, Bm, bias, Cm, M, K, Nc, relu);
  };
  auto fill = [&](float* p, float v, long n) {
    int blocks = (int)cdivl(n, TB); if (blocks > 8192) blocks = 8192;
    fill_f32<<<blocks, TB, 0, stream>>>(p, v, n);
  };

  // GAT layer: hin -> agg ([Nn, H*Ch] if concat else [Nn, Ch] head-mean)
  auto run_gat = [&](const float* hin, int layer, int Ch, bool concat) {
    const int NcW = H * Ch;
    launch_gemm(hin, W[layer], nullptr, xb, Nn, HID, NcW, 0);                 // x = h @ W
    attn_logits<<<(int)cdivl((long)Nn * H, wavesPerBlock), TB, 0, stream>>>(
        xb, As[layer], Ad[layer], als, ald, Nn, H, Ch);
    fill(mbuf, -INFINITY, (long)Nn * H);
    fill(den, 0.0f, (long)Nn * H);
    const long eh = (long)Etot * H;
    edge_logit_max<<<(int)cdivl(eh, TB), TB, 0, stream>>>(ei, E, Nn, H, als, ald, ebuf, mbuf);
    edge_exp_sum  <<<(int)cdivl(eh, TB), TB, 0, stream>>>(ei, E, Nn, H, mbuf, ebuf, den);
    fill(agg, 0.0f, (long)Nn * HID);   // both cases produce Nn*HID values
    edge_aggregate<<<4096, TB, 0, stream>>>(ei, E, Nn, H, Ch, xb, ebuf, den, agg, concat ? 0 : 1);
  };

  auto run_bn = [&](const float* Xraw, const float* bias, const float* g, const float* be,
                    const float* resid, float* outp, int elu_res) {
    fill(stats, 0.0f, 2L * HID);
    bn_stats<<<128, HID, 0, stream>>>(Xraw, bias, stats, Nn, HID);
    bn_apply<<<(int)cdivl((long)Nn * HID, TB), TB, 0, stream>>>(
        Xraw, bias, g, be, stats, resid, outp, Nn, HID, elu_res);
  };

  // ---- pipeline ----
  // input projection: h0 = x @ Wp + bp   (K=236, Nc=256)
  launch_gemm(x_in, Wp, bp, h0, Nn, F, HID, 0);

  // GAT layer 0: in h0 -> hA
  run_gat(h0, 0, C, true);
  run_bn(agg, Bb[0], Gg[0], Be[0], h0, hA, 1);

  // GAT layer 1: in hA -> hB
  run_gat(hA, 1, C, true);
  run_bn(agg, Bb[1], Gg[1], Be[1], hA, hB, 1);

  // GAT layer 2: in hB -> hA (head-mean, BN only)
  run_gat(hB, 2, HID, false);
  run_bn(agg, Bb[2], Gg[2], Be[2], nullptr, hA, 0);

  // classifier
  concat2<<<(int)cdivl((long)Nn * 2 * HID, TB), TB, 0, stream>>>(hA, h0, hc, Nn, HID);
  launch_gemm(hc, cW1, cb1, z1, Nn, 2 * HID, HID, 1);    // 512 -> 256, relu
  launch_gemm(z1, cW2, cb2, z2, Nn, HID, HID2, 1);       // 256 -> 128, relu
  final_fc2<<<(int)cdivl(Nn, TB), TB, 0, stream>>>(z2, cW3, cb3, (float*)d_out, Nn, HID2);

  (void)n_in; (void)out_size; (void)ws_size;
}